// Device_27788438405691
// MI455X (gfx1250) — compile-verified
//
#include <hip/hip_runtime.h>

#define WSZ 32
#define LOG_BLK 11
#define BLK (1 << LOG_BLK)          // 2048-float staging block (8 KB)
#define NISSUE (BLK / (WSZ * 4))    // 16 x b128 async-load instructions per block

#if __has_builtin(__builtin_amdgcn_global_load_async_to_lds_b128) && \
    __has_builtin(__builtin_amdgcn_s_wait_asynccnt)
#define USE_ASYNC_LDS 1
#define AS1 __attribute__((address_space(1)))
#define AS3 __attribute__((address_space(3)))
typedef int v4i __attribute__((ext_vector_type(4)));
#else
#define USE_ASYNC_LDS 0
#endif

__device__ __forceinline__ unsigned ballot32(int p) {
#if __has_builtin(__builtin_amdgcn_ballot_w32)
  return __builtin_amdgcn_ballot_w32((bool)p);
#else
  return (unsigned)__ballot(p != 0);
#endif
}

// Single-wave sequential scan. Steady regions: 32 steps/iter via a shuffle scan
// over the saturating-add monoid. Packet-TX: all P steps in one shot (they only
// depend on the packet base). Wake-preset: folded into the event handler.
// Input trace double-buffered into LDS with async global->LDS DMA (ASYNCcnt).
__global__ __launch_bounds__(WSZ, 1)
void energy_policy_scan(const float* __restrict__ h,
                        const float* __restrict__ pL,
                        const float* __restrict__ pOH,
                        const float* __restrict__ pTH,
                        const int*   __restrict__ pP,
                        float* __restrict__ e_out,
                        float* __restrict__ a_out,
                        int T) {
  const int lane = (int)threadIdx.x;
  const float L  = pL[0];
  const float OH = pOH[0];
  const float TH = pTH[0];
  const int   P  = pP[0];
  const float MAXE    = 4.0f * TH;
  const float WAKE_TH = 5.0f * L + OH;
  const float SEND_TH = TH + 5.0f * L;

  if (T <= 0) return;
  if (lane == 0) { e_out[0] = 0.0f; a_out[0] = 0.0f; }

#if USE_ASYNC_LDS
  __shared__ float hbuf[2][BLK];
  const int use_lds = (T >= 64);
  const int maxblk  = (T - 1) >> LOG_BLK;
  int issued_hi = -1, ready_hi = -1;
  auto issue_block = [&](int b) {   // async-DMA one 8KB block of the trace into LDS
    const int base = b << LOG_BLK;
#pragma unroll
    for (int i = 0; i < NISSUE; ++i) {
      int idx = base + i * (WSZ * 4) + lane * 4;
      if (idx + 4 > T) idx = (T - 4) & ~3;       // clamp: aligned, in-bounds
      __builtin_amdgcn_global_load_async_to_lds_b128(
          (AS1 v4i*)(h + idx),
          (AS3 v4i*)(&hbuf[b & 1][i * (WSZ * 4) + lane * 4]),
          0, 0);
    }
  };
  auto ensure_ready = [&](int t0, int span) {    // blocks covering [t0, t0+span)
    if (!use_lds) return;
    const int bc = t0 >> LOG_BLK;
    const int bh = (t0 + span - 1) >> LOG_BLK;
    if (issued_hi < bc + 1 && bc + 1 <= maxblk) { issue_block(bc + 1); issued_hi = bc + 1; }
    if (ready_hi < bh) {
      if (issued_hi > bh) __builtin_amdgcn_s_wait_asynccnt(NISSUE);  // next block may stay in flight
      else                __builtin_amdgcn_s_wait_asynccnt(0);
      asm volatile("" ::: "memory");
      ready_hi = bh;
    }
  };
  if (use_lds) {
    issue_block(0); issued_hi = 0;
    if (maxblk >= 1) { issue_block(1); issued_hi = 1; }
  }
#define HV(idx) (use_lds ? hbuf[((idx) >> LOG_BLK) & 1][(idx) & (BLK - 1)] : h[(idx)])
#else
  auto ensure_ready = [&](int, int) {};
#define HV(idx) (h[(idx)])
#endif

  // lax.scan carry, replicated uniformly across the wave
  float e = 0.0f;
  int   on = 0;
  int   send_rem = 0;
  float send_base = 0.0f;
  int   preset = 0;
  float preset_val = 0.0f;
  int   done = 0;

  int t = 1;
  while (t < T) {
    if (done) {                       // after a break the trace stays zero
      for (int i = t + lane; i < T; i += WSZ) { e_out[i] = 0.0f; a_out[i] = 0.0f; }
      break;
    }
    if (!preset && send_rem <= 0) {
      // ---------- steady mode: up to 32 timesteps per iteration ----------
      const int n = (T - t < WSZ) ? (T - t) : WSZ;
      ensure_ready(t, n);
#if !USE_ASYNC_LDS
      if (t + 1024 + WSZ <= T) __builtin_prefetch(h + t + 1024 + lane, 0, 1);
#endif
      const float hv = (lane < n) ? HV(t + lane) : 0.0f;

      // per-step fn f(x)=clamp(x+A,B,C); Hillis-Steele scan of compositions
      float A = hv - L, B = 0.0f, C = MAXE;
#pragma unroll
      for (int d = 1; d < WSZ; d <<= 1) {
        const float A0 = __shfl_up(A, (unsigned)d, WSZ);
        const float B0 = __shfl_up(B, (unsigned)d, WSZ);
        const float C0 = __shfl_up(C, (unsigned)d, WSZ);
        if (lane >= d) {
          const float Bn = fminf(fmaxf(B0 + A, B), C);
          const float Cn = fminf(fmaxf(C0 + A, B), C);
          A = A0 + A; B = Bn; C = Cn;
        }
      }
      const float ei = fminf(fmaxf(e + A, B), C);   // candidate e at step t+lane

      int is_off = 0, is_send = 0, is_wake = 0;
      if (on) { is_off = (ei == 0.0f); is_send = (ei != 0.0f) && (ei >= SEND_TH); }
      else    { is_wake = (ei >= WAKE_TH); }
      const int ev = (lane < n) && (is_off | is_send | is_wake);
      const unsigned mask = ballot32(ev);
      const int j = mask ? (int)__builtin_ctz(mask) : n;
      const int accept = (j < n) ? (j + 1) : n;

      const float aval = (ev && is_send && (t + lane + P + 1 < T)) ? 1.0f : 0.0f;
      if (lane < accept) { e_out[t + lane] = ei; a_out[t + lane] = aval; }

      if (j < n) {
        const int   te      = t + j;
        const float ej      = __shfl(ei, j, WSZ);
        const int   ej_send = __shfl(is_send, j, WSZ);
        const int   ej_wake = __shfl(is_wake, j, WSZ);
        if (ej_wake) {
          if (te + 1 >= T) { done = 1; t = te + 1; }   // wake_break
          else {
            const float ep = ej - OH;                  // preset step (no decision)
            if (lane == 0) { e_out[te + 1] = ep; a_out[te + 1] = 0.0f; }
            on = 1; e = ep; t = te + 2;
          }
        } else if (ej_send) {
          if (te + P + 1 >= T) { done = 1; t = te + 1; }  // send_break
          else { send_rem = P; send_base = ej; e = ej; t = te + 1; }
        } else {
          on = 0; e = ej; t = te + 1;                  // battery hit zero while on
        }
      } else {
        e = __shfl(ei, n - 1, WSZ);
        t += n;
      }
    } else if (!preset && send_rem == P && P <= WSZ && t + P <= T) {
      // ---------- packet TX: all P steps at once (depend only on send_base) ----------
      ensure_ready(t, P);
      float et = 0.0f;
      if (lane < P) {
        const float hv = HV(t + lane);
        et = send_base - TH * (float)(lane + 1) / (float)P + hv;   // no clip (as in ref)
        e_out[t + lane] = et;
        a_out[t + lane] = 0.0f;
      }
      e = __shfl(et, P - 1, WSZ);
      send_rem = 0;
      t += P;
    } else {
      // ---------- generic serial step (fallback), exact JAX-scan mirror ----------
      const float hv = h[t];
      const float e_norm = fminf(fmaxf(e + hv - L, 0.0f), MAXE);
      int jj = P - send_rem;
      jj = jj < 0 ? 0 : (jj > P - 1 ? P - 1 : jj);
      const float lin_j  = TH * (float)(jj + 1) / (float)P;
      const float e_send = send_base - lin_j + hv;
      const float et = preset ? preset_val : (send_rem > 0 ? e_send : e_norm);
      const int decide = (!preset) && (send_rem <= 0);
      int send_rem_n = send_rem - 1; if (send_rem_n < 0) send_rem_n = 0;
      const int wake       = decide && !on && (et >= WAKE_TH);
      const int wake_break = wake && (t + 1 >= T);
      const int off_ev     = decide && on && (et == 0.0f);
      const int send_ev    = decide && on && (et != 0.0f) && (et >= SEND_TH);
      const int send_break = send_ev && (t + P + 1 >= T);
      const int send_ok    = send_ev && !send_break;
      if (lane == 0) { e_out[t] = et; a_out[t] = send_ok ? 1.0f : 0.0f; }
      on         = wake ? 1 : (off_ev ? 0 : on);
      done       = (wake_break || send_break) ? 1 : 0;
      preset     = (wake && !wake_break) ? 1 : 0;
      preset_val = preset ? (et - OH) : 0.0f;
      send_rem   = send_ok ? P : send_rem_n;
      send_base  = send_ok ? et : send_base;
      e = et;
      t += 1;
    }
  }
#undef HV
}

extern "C" void kernel_launch(void* const* d_in, const int* in_sizes, int n_in,
                              void* d_out, int out_size, void* d_ws, size_t ws_size,
                              hipStream_t stream) {
  const float* h  = (const float*)d_in[0];   // e_harvest, T floats
  const float* L  = (const float*)d_in[1];   // leakage_per_sample (scalar)
  const float* OH = (const float*)d_in[2];   // init_overhead (scalar)
  const float* TH = (const float*)d_in[3];   // thresh (scalar)
  const int*   P  = (const int*)d_in[4];     // packet_size (scalar)
  const int T = in_sizes[0];
  float* e_out = (float*)d_out;              // e_trace
  float* a_out = e_out + T;                  // actions (0.0 / 1.0)
  (void)n_in; (void)out_size; (void)d_ws; (void)ws_size;
  energy_policy_scan<<<1, WSZ, 0, stream>>>(h, L, OH, TH, P, e_out, a_out, T);
}